// STAttBlock_16518444221188
// MI455X (gfx1250) — compile-verified
//
#include <hip/hip_runtime.h>
#include <math.h>

typedef float v2f __attribute__((ext_vector_type(2)));
typedef float v8f __attribute__((ext_vector_type(8)));

#define B_   4
#define T_   12
#define N_   384
#define D_   64
#define MID_ 32
#define KH_  8
#define SCALE_ 0.35355339059327373f   // 1/sqrt(8)
#define EPS_ 1e-5f

static __device__ __forceinline__ v8f wmma_f32(v2f a, v2f b, v8f c) {
  // D = A(16x4 f32) * B(4x16 f32) + C(16x16 f32)
  return __builtin_amdgcn_wmma_f32_16x16x4_f32(false, a, false, b, (short)0, c,
                                               false, false);
}

// ---------------------------------------------------------------------------
// WMMA GEMM, 4 M-tiles (64 rows) x 1 N-tile per wave so the B fragment is
// loaded once per k-chunk and reused by 4 WMMAs.
// MODE 0: A[m][k] = A0[m*K + k]
// MODE 1: A = concat([X|STE]) (K=128, two 64-col segments)
// MODE 2: TCN conv gather: p0=Tin, p1=ksize, 32 channels, spatial N_=384.
// ---------------------------------------------------------------------------
template<int MODE>
__global__ void gemm4_k(const float* __restrict__ A0, const float* __restrict__ A1,
                        const float* __restrict__ Bw, int ldb,
                        const float* __restrict__ bias,
                        float* __restrict__ C, int ldc,
                        int M, int N, int K, int act, int p0, int p1)
{
  (void)A1; (void)p0; (void)p1;
  int wid = (blockIdx.x * blockDim.x + threadIdx.x) >> 5;
  int ntn = N >> 4;
  int tiles = (M >> 6) * ntn;
  if (wid >= tiles) return;                 // wave-uniform exit
  int tmg = wid / ntn, tn = wid - tmg * ntn;
  int lane = threadIdx.x & 31;
  int half = lane >> 4, l16 = lane & 15;
  int col = tn * 16 + l16;
  int row[4];
#pragma unroll
  for (int i = 0; i < 4; ++i) row[i] = tmg * 64 + i * 16 + l16;

  v8f acc0 = {}, acc1 = {}, acc2 = {}, acc3 = {};

  if (MODE == 2) {
    int Tout = p0 - p1 + 1;
    int bb_[4], to_[4], nn_[4];
#pragma unroll
    for (int i = 0; i < 4; ++i) {
      bb_[i] = row[i] / (Tout * N_);
      int rem = row[i] - bb_[i] * Tout * N_;
      to_[i] = rem / N_; nn_[i] = rem - to_[i] * N_;
    }
    for (int dt = 0; dt < p1; ++dt) {
      const float* bp0 = A0 + (((size_t)bb_[0] * p0 + to_[0] + dt) * N_ + nn_[0]) * 32;
      const float* bp1 = A0 + (((size_t)bb_[1] * p0 + to_[1] + dt) * N_ + nn_[1]) * 32;
      const float* bp2 = A0 + (((size_t)bb_[2] * p0 + to_[2] + dt) * N_ + nn_[2]) * 32;
      const float* bp3 = A0 + (((size_t)bb_[3] * p0 + to_[3] + dt) * N_ + nn_[3]) * 32;
#pragma unroll
      for (int ci = 0; ci < 32; ci += 4) {
        int co = ci + half * 2;
        int ka = dt * 32 + co;
        v2f b, a;
        b.x = Bw[(size_t)ka * ldb + col];
        b.y = Bw[(size_t)(ka + 1) * ldb + col];
        a.x = bp0[co]; a.y = bp0[co + 1]; acc0 = wmma_f32(a, b, acc0);
        a.x = bp1[co]; a.y = bp1[co + 1]; acc1 = wmma_f32(a, b, acc1);
        a.x = bp2[co]; a.y = bp2[co + 1]; acc2 = wmma_f32(a, b, acc2);
        a.x = bp3[co]; a.y = bp3[co + 1]; acc3 = wmma_f32(a, b, acc3);
      }
    }
  } else {
    const int segs = (MODE == 1) ? 2 : 1;
    for (int sgm = 0; sgm < segs; ++sgm) {
      const float* As = (MODE == 1) ? (sgm ? A1 : A0) : A0;
      const int lda  = (MODE == 1) ? 64 : K;
      const int Kseg = (MODE == 1) ? 64 : K;
      const int kbase = sgm * 64;
      const float* r0p = As + (size_t)row[0] * lda;
      const float* r1p = As + (size_t)row[1] * lda;
      const float* r2p = As + (size_t)row[2] * lda;
      const float* r3p = As + (size_t)row[3] * lda;
      for (int k = 0; k < Kseg; k += 4) {
        int co = k + half * 2;
        int ka = kbase + co;
        v2f b, a;
        b.x = Bw[(size_t)ka * ldb + col];
        b.y = Bw[(size_t)(ka + 1) * ldb + col];
        a.x = r0p[co]; a.y = r0p[co + 1]; acc0 = wmma_f32(a, b, acc0);
        a.x = r1p[co]; a.y = r1p[co + 1]; acc1 = wmma_f32(a, b, acc1);
        a.x = r2p[co]; a.y = r2p[co + 1]; acc2 = wmma_f32(a, b, acc2);
        a.x = r3p[co]; a.y = r3p[co + 1]; acc3 = wmma_f32(a, b, acc3);
      }
    }
  }

  float bv = bias ? bias[col] : 0.0f;
  v8f accs[4] = {acc0, acc1, acc2, acc3};
#pragma unroll
  for (int i = 0; i < 4; ++i) {
#pragma unroll
    for (int r = 0; r < 8; ++r) {
      float x = accs[i][r] + bv;
      if (act == 1) x = fmaxf(x, 0.0f);
      C[(size_t)(tmg * 64 + i * 16 + r + half * 8) * ldc + col] = x;
    }
  }
}

// ---------------------------------------------------------------------------
// HS_pre[:,t] = mod_b[t] + sum_j O_buf[pair(t,j)] @ mod_W[t][j*64:(j+1)*64, :]
// O_buf: [pair][B][N][64] ; HS_pre: [B,T,N,D].  4 M-tiles per wave.
// ---------------------------------------------------------------------------
struct ModArgs { const float* W[12]; const float* bias[12]; };

__global__ void gemm_mod_k(const float* __restrict__ Obuf, ModArgs ma,
                           float* __restrict__ HSp)
{
  int t = blockIdx.y;
  int wid = (blockIdx.x * blockDim.x + threadIdx.x) >> 5;
  const int ntn = 4;                    // 64/16
  const int tiles = (1536 / 64) * ntn;  // 96
  if (wid >= tiles) return;
  int tmg = wid / ntn, tn = wid - tmg * ntn;
  int lane = threadIdx.x & 31, half = lane >> 4, l16 = lane & 15;
  int col = tn * 16 + l16;
  int pb = t * (t + 1) / 2;
  const float* Wt = ma.W[t];
  v8f acc0 = {}, acc1 = {}, acc2 = {}, acc3 = {};
  for (int j = 0; j <= t; ++j) {
    const float* A = Obuf + (size_t)(pb + j) * 1536 * 64;
    const float* r0p = A + (size_t)(tmg * 64 +  0 + l16) * 64;
    const float* r1p = A + (size_t)(tmg * 64 + 16 + l16) * 64;
    const float* r2p = A + (size_t)(tmg * 64 + 32 + l16) * 64;
    const float* r3p = A + (size_t)(tmg * 64 + 48 + l16) * 64;
    const float* Wj = Wt + (size_t)j * 64 * 64;
#pragma unroll
    for (int k = 0; k < 64; k += 4) {
      int co = k + half * 2;
      v2f b, a;
      b.x = Wj[(size_t)co * 64 + col];
      b.y = Wj[(size_t)(co + 1) * 64 + col];
      a.x = r0p[co]; a.y = r0p[co + 1]; acc0 = wmma_f32(a, b, acc0);
      a.x = r1p[co]; a.y = r1p[co + 1]; acc1 = wmma_f32(a, b, acc1);
      a.x = r2p[co]; a.y = r2p[co + 1]; acc2 = wmma_f32(a, b, acc2);
      a.x = r3p[co]; a.y = r3p[co + 1]; acc3 = wmma_f32(a, b, acc3);
    }
  }
  float bv = ma.bias[t][col];
  v8f accs[4] = {acc0, acc1, acc2, acc3};
#pragma unroll
  for (int i = 0; i < 4; ++i) {
#pragma unroll
    for (int r = 0; r < 8; ++r) {
      int m = tmg * 64 + i * 16 + r + half * 8;
      int b_ = m / N_, n = m - b_ * N_;
      HSp[(((size_t)b_ * T_ + t) * N_ + n) * D_ + col] = accs[i][r] + bv;
    }
  }
}

// ---------------------------------------------------------------------------
// Attention: grid (78 pairs, B, 4); block = 64 (2 waves, 1 head each).
// Per (pair p=(t,j), b, h): S = Qt Qj^T * SCALE, softmax rows, O = P Qj.
// P strip (16x384) per wave in LDS.  O_buf[pair][b][n][h*8+k].
// ---------------------------------------------------------------------------
__global__ void attn_k(const float* __restrict__ q, float* __restrict__ Obuf)
{
  __shared__ float Pl[2][16 * 384];   // 48 KB
  int p = blockIdx.x, b = blockIdx.y;
  int wv = threadIdx.x >> 5;
  int h = blockIdx.z * 2 + wv;
  int t = 0, accp = 0;
  while (accp + t + 1 <= p) { accp += t + 1; ++t; }
  int j = p - accp;
  int lane = threadIdx.x & 31, half = lane >> 4, l16 = lane & 15;
  const float* Qt = q + ((size_t)b * T_ + t) * N_ * D_ + h * KH_;
  const float* Qj = q + ((size_t)b * T_ + j) * N_ * D_ + h * KH_;
  float* P = Pl[wv];
  float* Op = Obuf + ((size_t)p * B_ + b) * N_ * D_ + h * KH_;

  for (int s = 0; s < 24; ++s) {
    int r0 = s * 16;
    int arow = r0 + l16;
    v2f a0, a1;
    a0.x = Qt[(size_t)arow * D_ + half * 2];
    a0.y = Qt[(size_t)arow * D_ + half * 2 + 1];
    a1.x = Qt[(size_t)arow * D_ + 4 + half * 2];
    a1.y = Qt[(size_t)arow * D_ + 4 + half * 2 + 1];
    // ---- scores ----
    for (int mt = 0; mt < 24; ++mt) {
      int m0 = mt * 16;
      int brow = m0 + l16;
      v2f b0, b1;
      b0.x = Qj[(size_t)brow * D_ + half * 2];
      b0.y = Qj[(size_t)brow * D_ + half * 2 + 1];
      b1.x = Qj[(size_t)brow * D_ + 4 + half * 2];
      b1.y = Qj[(size_t)brow * D_ + 4 + half * 2 + 1];
      v8f c = {};
      c = wmma_f32(a0, b0, c);
      c = wmma_f32(a1, b1, c);
#pragma unroll
      for (int r = 0; r < 8; ++r)
        P[(r + half * 8) * 384 + m0 + l16] = c[r] * SCALE_;
    }
    __syncthreads();
    // ---- softmax over 384 cols; 2 lanes per row ----
    {
      float* pr = P + l16 * 384 + half * 192;
      float mx = -3.0e38f;
      for (int i = 0; i < 192; ++i) mx = fmaxf(mx, pr[i]);
      mx = fmaxf(mx, __shfl_xor(mx, 16, 32));
      float sum = 0.f;
      for (int i = 0; i < 192; ++i) { float e = __expf(pr[i] - mx); pr[i] = e; sum += e; }
      sum += __shfl_xor(sum, 16, 32);
      float inv = 1.0f / sum;
      for (int i = 0; i < 192; ++i) pr[i] *= inv;
    }
    __syncthreads();
    // ---- O = P @ V (V cols padded 8->16) ----
    v8f o = {};
    for (int mt = 0; mt < 24; ++mt) {
#pragma unroll
      for (int sub = 0; sub < 4; ++sub) {
        int kb = mt * 16 + sub * 4;
        v2f a, bb;
        a.x = P[l16 * 384 + kb + half * 2];
        a.y = P[l16 * 384 + kb + half * 2 + 1];
        int kr = kb + half * 2;
        bb.x = (l16 < KH_) ? Qj[(size_t)kr * D_ + l16] : 0.0f;
        bb.y = (l16 < KH_) ? Qj[(size_t)(kr + 1) * D_ + l16] : 0.0f;
        o = wmma_f32(a, bb, o);
      }
    }
    if (l16 < KH_) {
#pragma unroll
      for (int r = 0; r < 8; ++r)
        Op[(size_t)(r0 + r + half * 8) * D_ + l16] = o[r];
    }
    __syncthreads();
  }
}

// ---------------------------------------------------------------------------
// BN statistics / elementwise helpers
// ---------------------------------------------------------------------------
__global__ void bn_stats_chan(const float* __restrict__ x, int rows, int C,
                              float* __restrict__ mean, float* __restrict__ rstd)
{
  int c = blockIdx.x, tid = threadIdx.x;
  float s1 = 0.f, s2 = 0.f;
  for (int r = tid; r < rows; r += blockDim.x) {
    float v = x[(size_t)r * C + c]; s1 += v; s2 += v * v;
  }
  __shared__ float a1[256], a2[256];
  a1[tid] = s1; a2[tid] = s2; __syncthreads();
  for (int s = 128; s > 0; s >>= 1) {
    if (tid < s) { a1[tid] += a1[tid + s]; a2[tid] += a2[tid + s]; }
    __syncthreads();
  }
  if (tid == 0) {
    float m = a1[0] / rows, v = a2[0] / rows - m * m;
    mean[c] = m; rstd[c] = rsqrtf(v + EPS_);
  }
}

__global__ void bn_stats_td(const float* __restrict__ x,
                            float* __restrict__ mean, float* __restrict__ rstd)
{
  int g = blockIdx.x, tid = threadIdx.x;    // g = t*64 + d, over 768 groups
  int t = g >> 6, d = g & 63;
  float s1 = 0.f, s2 = 0.f;
  for (int i = tid; i < B_ * N_; i += blockDim.x) {
    int b = i / N_, n = i - b * N_;
    float v = x[(((size_t)b * T_ + t) * N_ + n) * D_ + d];
    s1 += v; s2 += v * v;
  }
  __shared__ float a1[256], a2[256];
  a1[tid] = s1; a2[tid] = s2; __syncthreads();
  for (int s = 128; s > 0; s >>= 1) {
    if (tid < s) { a1[tid] += a1[tid + s]; a2[tid] += a2[tid + s]; }
    __syncthreads();
  }
  if (tid == 0) {
    float m = a1[0] / (B_ * N_), v = a2[0] / (B_ * N_) - m * m;
    mean[g] = m; rstd[g] = rsqrtf(v + EPS_);
  }
}

__global__ void apply_td_relu(const float* __restrict__ x, float* __restrict__ y,
                              const float* __restrict__ mean, const float* __restrict__ rstd,
                              const float* __restrict__ g, const float* __restrict__ bb,
                              int total)
{
  int idx = blockIdx.x * blockDim.x + threadIdx.x;
  if (idx >= total) return;
  int d = idx & 63;
  int t = ((idx >> 6) / N_) % T_;
  int grp = t * 64 + d;
  float v = (x[idx] - mean[grp]) * rstd[grp] * g[d] + bb[d];
  y[idx] = fmaxf(v, 0.0f);
}

__global__ void apply_chan(const float* __restrict__ x, float* __restrict__ y,
                           const float* __restrict__ mean, const float* __restrict__ rstd,
                           const float* __restrict__ g, const float* __restrict__ bb,
                           int total, int C, int relu)
{
  int idx = blockIdx.x * blockDim.x + threadIdx.x;
  if (idx >= total) return;
  int c = idx % C;
  float v = (x[idx] - mean[c]) * rstd[c] * g[c] + bb[c];
  y[idx] = relu ? fmaxf(v, 0.0f) : v;
}

__global__ void tcn_act_k(const float* __restrict__ a, const float* __restrict__ g,
                          float* __restrict__ o, int total)
{
  int idx = blockIdx.x * blockDim.x + threadIdx.x;
  if (idx >= total) return;
  float av = fmaxf(a[idx], 0.0f);
  float gv = 1.0f / (1.0f + __expf(-g[idx]));
  o[idx] = av * gv;
}

__global__ void tcn_finish_k(const float* __restrict__ act,
                             const float* __restrict__ mean, const float* __restrict__ rstd,
                             const float* __restrict__ g, const float* __restrict__ bb,
                             const float* __restrict__ r1, int r1Tin, int r1off,
                             const float* __restrict__ r2, int r2off,
                             float* __restrict__ out, int Tout, int total)
{
  int idx = blockIdx.x * blockDim.x + threadIdx.x;
  if (idx >= total) return;
  int c = idx & 31;
  int tmp = idx >> 5;
  int n = tmp % N_;
  int t2 = tmp / N_;
  int to = t2 % Tout;
  int b = t2 / Tout;
  float v = (act[idx] - mean[c]) * rstd[c] * g[c] + bb[c];
  v += r1[(((size_t)b * r1Tin + to + r1off) * N_ + n) * 32 + c];
  if (r2) v += r2[(((size_t)b * T_ + to + r2off) * N_ + n) * 32 + c];
  out[idx] = v;
}

__global__ void colsum_k(const float* __restrict__ W, float* __restrict__ cs)
{
  int d = threadIdx.x;   // 64 threads
  float s = 0.f;
  for (int r = 0; r < 32; ++r) s += W[r * 64 + d];
  cs[d] = s;
}

__global__ void expand_HT_k(const float* __restrict__ U, const float* __restrict__ cs,
                            const float* __restrict__ inW, const float* __restrict__ inb,
                            const float* __restrict__ l2b, float* __restrict__ HT,
                            int total)
{
  int idx = blockIdx.x * blockDim.x + threadIdx.x;
  if (idx >= total) return;
  int d = idx & 63;
  int nd = idx >> 6;
  int n = nd % N_;
  int bt = nd / N_;
  int t = bt % T_, b = bt / T_;
  int m = b * N_ + n;
  HT[idx] = inW[t] * U[(size_t)m * 64 + d] + inb[t] * cs[d] + l2b[d];
}

__global__ void gate_k(const float* __restrict__ XSp, const float* __restrict__ XTp,
                       const float* __restrict__ msx, const float* __restrict__ rsx,
                       const float* __restrict__ gxs, const float* __restrict__ bxs,
                       const float* __restrict__ mxt, const float* __restrict__ rxt,
                       const float* __restrict__ gxt, const float* __restrict__ bxt,
                       const float* __restrict__ HS, const float* __restrict__ HT,
                       float* __restrict__ H, int total)
{
  int idx = blockIdx.x * blockDim.x + threadIdx.x;
  if (idx >= total) return;
  int c = idx & 63;
  float xs = (XSp[idx] - msx[c]) * rsx[c] * gxs[c] + bxs[c];
  float xt = (XTp[idx] - mxt[c]) * rxt[c] * gxt[c] + bxt[c];
  float z = 1.0f / (1.0f + __expf(-(xs + xt)));
  H[idx] = z * HS[idx] + (1.0f - z) * HT[idx];
}

__global__ void final_k(const float* __restrict__ X, const float* __restrict__ e,
                        const float* __restrict__ mean, const float* __restrict__ rstd,
                        const float* __restrict__ g, const float* __restrict__ bb,
                        float* __restrict__ out, int total)
{
  int idx = blockIdx.x * blockDim.x + threadIdx.x;
  if (idx >= total) return;
  int c = idx & 63;
  out[idx] = X[idx] + (e[idx] - mean[c]) * rstd[c] * g[c] + bb[c];
}

// ---------------------------------------------------------------------------
// Host driver
// ---------------------------------------------------------------------------
static inline int gemm_blocks4(int M, int N) {
  int waves = (M >> 6) * (N >> 4);
  return (waves + 7) / 8;
}

extern "C" void kernel_launch(void* const* d_in, const int* in_sizes, int n_in,
                              void* d_out, int out_size, void* d_ws, size_t ws_size,
                              hipStream_t stream)
{
  (void)in_sizes; (void)n_in; (void)out_size;
  const float* const* in = (const float* const*)d_in;
  const float* X   = in[0];
  const float* STE = in[1];

  // workspace arena (floats)
  const size_t OFF_Q = 0,         SZ_BTND = 1179648;
  const size_t OFF_RAW = 1179648;
  const size_t OFF_OBUF = 2359296;               // 78*4*384*64 = 7,667,712
  const size_t OFF_HSP = 10027008;
  const size_t OFF_HS  = 11206656;
  const size_t OFF_H   = 12386304;               // 589,824
  const size_t OFF_TA  = 12976128, OFF_TG = 13467648, OFF_TACT = 13959168;
  const size_t OFF_TO0 = 14450688, OFF_TO1 = 14942208;
  const size_t OFF_VB  = 15433728, OFF_U = 15482880;
  const size_t OFF_HT  = 15581184;
  const size_t OFF_HB  = 16760832;
  const size_t OFF_CS  = 17940480;
  const size_t OFF_ST  = 17940544;
  const size_t WS_TOTAL = 17944640;
  if (ws_size < WS_TOTAL * sizeof(float)) return;

  float* ws = (float*)d_ws;
  float* q    = ws + OFF_Q;
  float* raw  = ws + OFF_RAW;     // qkv pre-BN, later reused for safc pre-BN
  float* Obuf = ws + OFF_OBUF;
  float* HSp  = ws + OFF_HSP;
  float* HS   = ws + OFF_HS;
  float* h    = ws + OFF_H;
  float* ta = ws + OFF_TA, *tg = ws + OFF_TG, *tact = ws + OFF_TACT;
  float* to0 = ws + OFF_TO0, *to1 = ws + OFF_TO1;
  float* vb = ws + OFF_VB, *U = ws + OFF_U;
  float* HT = ws + OFF_HT, *Hb = ws + OFF_HB;
  float* cs = ws + OFF_CS;
  float* st = ws + OFF_ST;
  float* td_m = st, *td_r = st + 768;
  float* sSA_m = st + 1536, *sSA_r = st + 1600;   // safc
  float* sTC_m = st + 1664, *sTC_r = st + 1728;   // tcn (reused per layer)
  float* sXS_m = st + 1792, *sXS_r = st + 1856;
  float* sXT_m = st + 1920, *sXT_r = st + 1984;
  float* sB1_m = st + 2048, *sB1_r = st + 2112;
  float* sB2_m = st + 2176, *sB2_r = st + 2240;
  // gated-fusion temporaries alias the (dead) attention O buffer
  float* XSp = Obuf;
  float* XTp = Obuf + SZ_BTND;
  float* t1 = Obuf + 2 * SZ_BTND, *t2 = Obuf + 3 * SZ_BTND, *t3 = Obuf + 4 * SZ_BTND;

  const int BTND = B_ * T_ * N_ * D_;   // 1,179,648
  const int EW = 256;
  #define EWB(n) (((n) + EW - 1) / EW)

  // 1) qkv = [X|STE] @ qkv_W + qkv_b
  gemm4_k<1><<<gemm_blocks4(18432, 64), 256, 0, stream>>>(
      X, STE, in[2], 64, in[3], raw, 64, 18432, 64, 128, 0, 0, 0);
  // 2) BN over (B,N) per (t,d); relu -> q
  bn_stats_td<<<768, 256, 0, stream>>>(raw, td_m, td_r);
  apply_td_relu<<<EWB(BTND), EW, 0, stream>>>(raw, q, td_m, td_r, in[4], in[5], BTND);
  // 3) attention for all 78 (t,j) pairs
  attn_k<<<dim3(78, B_, 4), 64, 0, stream>>>(q, Obuf);
  // 4) per-t reduction through mod_W
  ModArgs ma;
  for (int t = 0; t < 12; ++t) { ma.W[t] = in[10 + t]; ma.bias[t] = in[22 + t]; }
  gemm_mod_k<<<dim3(12, 12), 256, 0, stream>>>(Obuf, ma, HSp);
  // 5) safc + BN + relu -> HS
  gemm4_k<0><<<gemm_blocks4(18432, 64), 256, 0, stream>>>(
      HSp, nullptr, in[6], 64, in[7], raw, 64, 18432, 64, 64, 0, 0, 0);
  bn_stats_chan<<<64, 256, 0, stream>>>(raw, 18432, 64, sSA_m, sSA_r);
  apply_chan<<<EWB(BTND), EW, 0, stream>>>(raw, HS, sSA_m, sSA_r, in[8], in[9], BTND, 64, 1);
  // 6) h = [X|STE] @ tlin_W + tlin_b
  gemm4_k<1><<<gemm_blocks4(18432, 32), 256, 0, stream>>>(
      X, STE, in[34], 32, in[35], h, 32, 18432, 32, 128, 0, 0, 0);
  // 7) TCN stack
  const int Tin[6] = {12, 10, 8, 6, 4, 2}, KS[6] = {3, 3, 3, 3, 3, 2};
  const float* cur = h;
  float* outs[2] = {to0, to1};
  for (int i = 0; i < 6; ++i) {
    int k = KS[i], Ti = Tin[i], To = Ti - k + 1;
    int M = B_ * To * N_, K = k * 32, tot = M * 32;
    const float* W1 = in[36 + 6 * i], *b1 = in[37 + 6 * i];
    const float* W2 = in[38 + 6 * i], *b2 = in[39 + 6 * i];
    const float* bg = in[40 + 6 * i], *bbp = in[41 + 6 * i];
    gemm4_k<2><<<gemm_blocks4(M, 32), 256, 0, stream>>>(
        cur, nullptr, W1, 32, b1, ta, 32, M, 32, K, 0, Ti, k);
    gemm4_k<2><<<gemm_blocks4(M, 32), 256, 0, stream>>>(
        cur, nullptr, W2, 32, b2, tg, 32, M, 32, K, 0, Ti, k);
    tcn_act_k<<<EWB(tot), EW, 0, stream>>>(ta, tg, tact, tot);
    bn_stats_chan<<<32, 256, 0, stream>>>(tact, M, 32, sTC_m, sTC_r);
    float* outp = outs[i & 1];
    int r1off = (i == 5) ? 1 : 2;
    const float* r2 = (i == 0) ? nullptr : h;
    int r2off = (i == 5) ? 11 : 2 * (i + 1);
    tcn_finish_k<<<EWB(tot), EW, 0, stream>>>(
        tact, sTC_m, sTC_r, bg, bbp, cur, Ti, r1off, r2, r2off, outp, To, tot);
    cur = outp;
  }
  // 8) v = relu(out0 @ fully_W + b); U = v @ lin2_W; HT expand
  gemm4_k<0><<<gemm_blocks4(1536, 32), 256, 0, stream>>>(
      cur, nullptr, in[72], 32, in[73], vb, 32, 1536, 32, 32, 1, 0, 0);
  gemm4_k<0><<<gemm_blocks4(1536, 64), 256, 0, stream>>>(
      vb, nullptr, in[76], 64, nullptr, U, 64, 1536, 64, 32, 0, 0, 0);
  colsum_k<<<1, 64, 0, stream>>>(in[76], cs);
  expand_HT_k<<<EWB(BTND), EW, 0, stream>>>(U, cs, in[74], in[75], in[77], HT, BTND);
  // 9) gated fusion
  gemm4_k<0><<<gemm_blocks4(18432, 64), 256, 0, stream>>>(
      HS, nullptr, in[78], 64, nullptr, XSp, 64, 18432, 64, 64, 0, 0, 0);
  gemm4_k<0><<<gemm_blocks4(18432, 64), 256, 0, stream>>>(
      HT, nullptr, in[81], 64, in[82], XTp, 64, 18432, 64, 64, 0, 0, 0);
  bn_stats_chan<<<64, 256, 0, stream>>>(XSp, 18432, 64, sXS_m, sXS_r);
  bn_stats_chan<<<64, 256, 0, stream>>>(XTp, 18432, 64, sXT_m, sXT_r);
  gate_k<<<EWB(BTND), EW, 0, stream>>>(XSp, XTp, sXS_m, sXS_r, in[79], in[80],
                                       sXT_m, sXT_r, in[83], in[84], HS, HT, Hb, BTND);
  gemm4_k<0><<<gemm_blocks4(18432, 64), 256, 0, stream>>>(
      Hb, nullptr, in[85], 64, in[86], t1, 64, 18432, 64, 64, 0, 0, 0);
  bn_stats_chan<<<64, 256, 0, stream>>>(t1, 18432, 64, sB1_m, sB1_r);
  apply_chan<<<EWB(BTND), EW, 0, stream>>>(t1, t2, sB1_m, sB1_r, in[87], in[88], BTND, 64, 1);
  gemm4_k<0><<<gemm_blocks4(18432, 64), 256, 0, stream>>>(
      t2, nullptr, in[89], 64, in[90], t3, 64, 18432, 64, 64, 0, 0, 0);
  bn_stats_chan<<<64, 256, 0, stream>>>(t3, 18432, 64, sB2_m, sB2_r);
  final_k<<<EWB(BTND), EW, 0, stream>>>(X, t3, sB2_m, sB2_r, in[91], in[92],
                                        (float*)d_out, BTND);
}